// Deriv2Matern52_59794534695229
// MI455X (gfx1250) — compile-verified
//
#include <hip/hip_runtime.h>
#include <hip/hip_bf16.h>

typedef __attribute__((ext_vector_type(2))) float v2f;
typedef __attribute__((ext_vector_type(4))) float v4f;
typedef __attribute__((ext_vector_type(8))) float v8f;

#define SQRT5F 2.23606797749978969f

// Each wave handles a 16(i) x 16(j) tile.
// Block = 256 threads = 8 waves => block tile = 16(i) x 128(j).
__global__ __launch_bounds__(256) void matern52_hess_kernel(
    const float* __restrict__ X1, const float* __restrict__ X2,
    const float* __restrict__ cptr, const float* __restrict__ lptr,
    float* __restrict__ out, int n, int m)
{
    const int jGroups = (m + 127) >> 7;
    const int it = blockIdx.x / jGroups;      // i-tile index (16 rows)
    const int jg = blockIdx.x % jGroups;      // j-group index (128 cols)
    const int i0 = it << 4;
    const int jBlock = jg << 7;

    // Pre-scaled tiles Y = X / l. Row stride 12 floats = 48B:
    // 16B-aligned rows (b128-friendly) and conflict-free across 16 lanes.
    __shared__ float sY1[16][12];
    __shared__ float sY2[128][12];

    for (int idx = threadIdx.x; idx < 16 * 8; idx += 256) {
        int r = idx >> 3, a = idx & 7;
        int gi = i0 + r;
        float x = (gi < n) ? X1[gi * 8 + a] : 0.0f;
        sY1[r][a] = x * __builtin_amdgcn_rcpf(lptr[a]);   // v_rcp_f32 + v_mul
    }
    for (int idx = threadIdx.x; idx < 128 * 8; idx += 256) {
        int r = idx >> 3, a = idx & 7;
        int gj = jBlock + r;
        float x = (gj < m) ? X2[gj * 8 + a] : 0.0f;
        sY2[r][a] = x * __builtin_amdgcn_rcpf(lptr[a]);
    }

    // Per-thread uniform constants (uniform addresses -> scalar ops).
    // Only constant-indexed uses below (fully unrolled) -> no cndmask trees.
    float il[8], cil2[8];
    const float cc = cptr[0] * cptr[0];
#pragma unroll
    for (int a = 0; a < 8; ++a) {
        il[a]   = __builtin_amdgcn_rcpf(lptr[a]);
        cil2[a] = cc * il[a] * il[a];
    }

    __syncthreads();

    const int lane = threadIdx.x & 31;
    const int wave = threadIdx.x >> 5;
    const int half = lane >> 4;               // 0: lanes 0-15, 1: lanes 16-31
    const int MN   = lane & 15;               // M for A-operand, N (=j col) for B/C/D
    const int jloc0 = wave << 4;              // this wave's j sub-tile inside sY2
    const int j  = jBlock + jloc0 + MN;       // global j column owned by this lane

    // ---- Gram matrix G = Y1 . Y2^T via V_WMMA_F32_16X16X4_F32 (K=8 in 2 steps) ----
    // f32 A 16x4 layout: lanes 0-15 -> {K=k0, k0+1}, lanes 16-31 -> {K=k0+2, k0+3}
    // f32 B 4x16 mirrors it per lane-half; C/D: VGPR v -> M = v + 8*half, N = lane&15
    v8f acc = {0.f, 0.f, 0.f, 0.f, 0.f, 0.f, 0.f, 0.f};
#pragma unroll
    for (int k0 = 0; k0 < 8; k0 += 4) {
        const int ka = k0 + (half << 1);      // even -> 8B-aligned ds_load_b64
        v2f A = { sY1[MN][ka],         sY1[MN][ka + 1] };
        v2f B = { sY2[jloc0 + MN][ka], sY2[jloc0 + MN][ka + 1] };
        acc = __builtin_amdgcn_wmma_f32_16x16x4_f32(
            /*neg_a=*/false, A, /*neg_b=*/false, B,
            /*c_mod=*/(short)0, acc, /*reuse_a=*/false, /*reuse_b=*/false);
    }

    // ---- Per-lane column data: y2 row, ||Y2_j||^2 ----
    float y2v[8];
    float n2 = 0.0f;
#pragma unroll
    for (int a = 0; a < 8; ++a) {
        y2v[a] = sY2[jloc0 + MN][a];
        n2 = fmaf(y2v[a], y2v[a], n2);
    }

    const size_t ms = (size_t)m;
    const bool jok = (j < m);

    // ---- Finish 8 (i,j) pairs per lane; 64 outputs each ----
#pragma unroll
    for (int v = 0; v < 8; ++v) {
        const int M = v + (half << 3);        // row inside i-tile
        const int i = i0 + M;

        float Dv[8];
        float n1 = 0.0f;
#pragma unroll
        for (int a = 0; a < 8; ++a) {
            float y1 = sY1[M][a];
            n1 = fmaf(y1, y1, n1);
            Dv[a] = (y1 - y2v[a]) * il[a];    // == (x1-x2)/l^2
        }

        float r2 = n1 + n2 - 2.0f * acc[v];
        r2 = fmaxf(r2, 0.0f);
        float r  = __builtin_amdgcn_sqrtf(r2);   // raw v_sqrt_f32, no fixup
        float e  = __expf(-SQRT5F * r);
        float fr = (5.0f / 3.0f) * e;
        float Af = fr * fmaf(SQRT5F, r, 1.0f);

        // vals[b] = Dv[a] * Ds[b]; diagonal adds Af*cc/l[a]^2.
        const float t = -5.0f * fr * cc;
        float Ds[8];
#pragma unroll
        for (int b = 0; b < 8; ++b) Ds[b] = t * Dv[b];

        if (i < n && jok) {
            float* rowp = out + (((size_t)i * 8) * ms + (size_t)j) * 8;
#pragma unroll
            for (int a = 0; a < 8; ++a) {
                float vals[8];
#pragma unroll
                for (int b = 0; b < 8; ++b) vals[b] = Dv[a] * Ds[b];
                vals[a] = fmaf(Af, cil2[a], vals[a]);

                v4f* p = (v4f*)(rowp + (size_t)a * ms * 8);
                v4f lo = {vals[0], vals[1], vals[2], vals[3]};
                v4f hi = {vals[4], vals[5], vals[6], vals[7]};
                __builtin_nontemporal_store(lo, p);       // streaming 256MB out:
                __builtin_nontemporal_store(hi, p + 1);   // keep it out of L2
            }
        }
    }
}

extern "C" void kernel_launch(void* const* d_in, const int* in_sizes, int n_in,
                              void* d_out, int out_size, void* d_ws, size_t ws_size,
                              hipStream_t stream) {
    const float* X1 = (const float*)d_in[0];
    const float* X2 = (const float*)d_in[1];
    const float* c  = (const float*)d_in[2];
    const float* l  = (const float*)d_in[3];
    float* out = (float*)d_out;

    const int d = in_sizes[3];                // feature dim (8)
    const int n = in_sizes[0] / d;
    const int m = in_sizes[1] / d;

    const int iTiles  = (n + 15) >> 4;
    const int jGroups = (m + 127) >> 7;
    dim3 grid(iTiles * jGroups);
    matern52_hess_kernel<<<grid, 256, 0, stream>>>(X1, X2, c, l, out, n, m);
}